// Model_45079976739011
// MI455X (gfx1250) — compile-verified
//
#include <hip/hip_runtime.h>
#include <hip/hip_bf16.h>

typedef float v2f __attribute__((ext_vector_type(2)));
typedef float v8f __attribute__((ext_vector_type(8)));

#define NEG_SLOPE 0.2f
#define KDIM 64          // inner dim of every GEMM in this model (D == H == 64)

// ---------------------------------------------------------------------------
// WMMA f32 GEMM:  Y[M,Nout] = X[M,64] @ W[64,Nout] + bias
// blockDim = 128 (4 waves); wave w owns column tile w (uniform exit when
// n0 >= Nout). Each wave preloads its entire B column-tile (16 v2f frags,
// 32 VGPRs) and then sweeps MT=4 row tiles, so the k-loop is just
// global_load_b64(A) + v_wmma_f32_16x16x4_f32 chained on the accumulator.
//
// f32 16x16x4 VGPR layout (ISA 7.12.2):
//   A (16x4):  lanes 0-15 -> M=lane, {v0,v1} = K 0,1 ; lanes 16-31 -> K 2,3
//   B (4x16):  lanes 0-15 -> N=lane, {v0,v1} = K 0,1 ; lanes 16-31 -> K 2,3
//   C/D (16x16): lanes 0-15 cols, rows 0-7 in v0..7; lanes 16-31 rows 8-15
// ---------------------------------------------------------------------------
#define MT 4             // M tiles per wave

__global__ __launch_bounds__(128) void gemm_wmma_f32(
    const float* __restrict__ X, const float* __restrict__ W,
    const float* __restrict__ bias, float* __restrict__ Y,
    int Mtiles, int Nout)
{
    const int ntile = threadIdx.x >> 5;
    const int n0 = ntile << 4;
    if (n0 >= Nout) return;                 // uniform per wave
    const int lane = threadIdx.x & 31;
    const int rc   = lane & 15;             // row of A / col of B,D
    const int kh   = (lane >> 4) << 1;      // 0 or 2 (k sub-offset)

    // Preload B fragments for all 16 k-steps (K = 64)
    v2f bfrag[KDIM / 4];
    #pragma unroll
    for (int ks = 0; ks < KDIM / 4; ++ks) {
        bfrag[ks].x = W[(size_t)(4 * ks + kh)     * Nout + n0 + rc];
        bfrag[ks].y = W[(size_t)(4 * ks + kh + 1) * Nout + n0 + rc];
    }
    const float bb    = bias[n0 + rc];
    const int   rbase = (lane >> 4) << 3;   // rows 0..7 or 8..15

    const int mt0 = blockIdx.x * MT;
    #pragma unroll
    for (int t = 0; t < MT; ++t) {
        const int mtile = mt0 + t;
        if (mtile >= Mtiles) return;        // uniform per wave (tail)
        const int m0 = mtile << 4;
        const float* xrow = X + (size_t)(m0 + rc) * KDIM + kh;

        v8f acc = {0.f, 0.f, 0.f, 0.f, 0.f, 0.f, 0.f, 0.f};
        #pragma unroll
        for (int ks = 0; ks < KDIM / 4; ++ks) {
            v2f a = *(const v2f*)(xrow + 4 * ks);        // global_load_b64
            acc = __builtin_amdgcn_wmma_f32_16x16x4_f32(
                      false, a, false, bfrag[ks], (short)0, acc, false, false);
        }

        float* yout = Y + (size_t)(m0 + rbase) * Nout + n0 + rc;
        #pragma unroll
        for (int r = 0; r < 8; ++r)
            yout[(size_t)r * Nout] = acc[r] + bb;
    }
}

// ---------------------------------------------------------------------------
// float atomic max via signed-int max / unsigned-int min trick (m init -inf)
// ---------------------------------------------------------------------------
__device__ __forceinline__ void atomicMaxF(float* addr, float val) {
    if (!(val < 0.0f)) atomicMax((int*)addr, __float_as_int(val));
    else               atomicMin((unsigned int*)addr, __float_as_uint(val));
}

// zero agg[N*64], m = -inf, denom = 0
__global__ void init_layer(float* __restrict__ agg, float* __restrict__ m,
                           float* __restrict__ den, int N)
{
    int i = blockIdx.x * blockDim.x + threadIdx.x;
    int n64 = N << 6;
    if (i < n64) agg[i] = 0.0f;
    if (i < N) { m[i] = -INFINITY; den[i] = 0.0f; }
}

// wave32 per edge: e = leaky_relu(xl[src]+xr[dst]) . att ; atomic max m[dst]
// lane handles features {2*lane, 2*lane+1} -> one b64 load per operand
__global__ __launch_bounds__(256) void edge_scores(
    const float* __restrict__ xl, const float* __restrict__ xr,
    const int* __restrict__ src, const int* __restrict__ dst,
    const float* __restrict__ att, float* __restrict__ e,
    float* __restrict__ m, int E)
{
    int wid  = (blockIdx.x * blockDim.x + threadIdx.x) >> 5;
    if (wid >= E) return;                  // uniform per wave
    int lane = threadIdx.x & 31;
    int s = src[wid], d = dst[wid];
    v2f zl = ((const v2f*)(xl + ((size_t)s << 6)))[lane];
    v2f zr = ((const v2f*)(xr + ((size_t)d << 6)))[lane];
    v2f av = ((const v2f*)att)[lane];
    float z0 = zl.x + zr.x;
    float z1 = zl.y + zr.y;
    float l0 = z0 > 0.0f ? z0 : NEG_SLOPE * z0;
    float l1 = z1 > 0.0f ? z1 : NEG_SLOPE * z1;
    float p  = l0 * av.x + l1 * av.y;
    #pragma unroll
    for (int off = 16; off > 0; off >>= 1)
        p += __shfl_xor(p, off, 32);
    if (lane == 0) { e[wid] = p; atomicMaxF(&m[d], p); }
}

// thread per edge: e <- exp(e - m[dst]); denom[dst] += e
__global__ void edge_exp(float* __restrict__ e, const float* __restrict__ m,
                         float* __restrict__ den, const int* __restrict__ dst,
                         int E)
{
    int i = blockIdx.x * blockDim.x + threadIdx.x;
    if (i >= E) return;
    int d = dst[i];
    float v = __expf(e[i] - m[d]);
    e[i] = v;
    atomicAdd(&den[d], v);
}

// wave32 per edge: agg[dst] += (ex/denom[dst]) * xl[src]
__global__ __launch_bounds__(256) void edge_scatter(
    const float* __restrict__ xl, const int* __restrict__ src,
    const int* __restrict__ dst, const float* __restrict__ ex,
    const float* __restrict__ den, float* __restrict__ agg, int E)
{
    int wid  = (blockIdx.x * blockDim.x + threadIdx.x) >> 5;
    if (wid >= E) return;                  // uniform per wave
    int lane = threadIdx.x & 31;
    int s = src[wid], d = dst[wid];
    float alpha = ex[wid] / den[d];
    v2f xv = ((const v2f*)(xl + ((size_t)s << 6)))[lane];
    float* pa = agg + ((size_t)d << 6) + (lane << 1);
    atomicAdd(&pa[0], alpha * xv.x);
    atomicAdd(&pa[1], alpha * xv.y);
}

// h[i] = relu(agg[i] + bias[i & 63])  (in place)
__global__ void bias_relu(float* __restrict__ h, const float* __restrict__ bias,
                          int total)
{
    int i = blockIdx.x * blockDim.x + threadIdx.x;
    if (i >= total) return;
    float v = h[i] + bias[i & 63];
    h[i] = v > 0.0f ? v : 0.0f;
}

__global__ void pool_init(float* __restrict__ sums, float* __restrict__ cnt,
                          int G)
{
    int i = blockIdx.x * blockDim.x + threadIdx.x;
    int g64 = G << 6;
    if (i < g64) sums[i] = 0.0f;
    if (i < G)   cnt[i]  = 0.0f;
}

// wave32 per node: sums[batch[n]] += h[n]; cnt[batch[n]] += 1
__global__ __launch_bounds__(256) void pool_scatter(
    const float* __restrict__ h, const int* __restrict__ batch,
    float* __restrict__ sums, float* __restrict__ cnt, int N)
{
    int wid  = (blockIdx.x * blockDim.x + threadIdx.x) >> 5;
    if (wid >= N) return;                  // uniform per wave
    int lane = threadIdx.x & 31;
    int g = batch[wid];
    v2f hv = ((const v2f*)(h + ((size_t)wid << 6)))[lane];
    float* ps = sums + ((size_t)g << 6) + (lane << 1);
    atomicAdd(&ps[0], hv.x);
    atomicAdd(&ps[1], hv.y);
    if (lane == 0) atomicAdd(&cnt[g], 1.0f);
}

// sums[g,h] /= max(cnt[g], 1)  (in place -> pooled)
__global__ void pool_finalize(float* __restrict__ sums,
                              const float* __restrict__ cnt, int G)
{
    int i = blockIdx.x * blockDim.x + threadIdx.x;
    if (i >= (G << 6)) return;
    float c = cnt[i >> 6];
    sums[i] /= (c > 1.0f ? c : 1.0f);
}

// ---------------------------------------------------------------------------
extern "C" void kernel_launch(void* const* d_in, const int* in_sizes, int n_in,
                              void* d_out, int out_size, void* d_ws, size_t ws_size,
                              hipStream_t stream)
{
    const float* x     = (const float*)d_in[0];
    const int*   ei    = (const int*)  d_in[1];
    const int*   batch = (const int*)  d_in[2];
    const float* Wl1   = (const float*)d_in[3];
    const float* bl1   = (const float*)d_in[4];
    const float* Wr1   = (const float*)d_in[5];
    const float* br1   = (const float*)d_in[6];
    const float* att1  = (const float*)d_in[7];
    const float* bias1 = (const float*)d_in[8];
    const float* Wl2   = (const float*)d_in[9];
    const float* bl2   = (const float*)d_in[10];
    const float* Wr2   = (const float*)d_in[11];
    const float* br2   = (const float*)d_in[12];
    const float* att2  = (const float*)d_in[13];
    const float* bias2 = (const float*)d_in[14];
    const float* lin_w = (const float*)d_in[15];
    const float* lin_b = (const float*)d_in[16];

    const int N = in_sizes[0] / 64;     // 50000
    const int E = in_sizes[1] / 2;      // 800000
    const int G = out_size / 32;        // 512
    const int N64  = N << 6;
    const int Npad = (N + 63) & ~63;

    const int* src = ei;
    const int* dst = ei + E;

    // workspace layout (floats)
    float* buf0 = (float*)d_ws;          // xl        [N*64]
    float* buf1 = buf0 + N64;            // xr        [N*64]
    float* buf2 = buf1 + N64;            // agg / h   [N*64]
    float* e    = buf2 + N64;            // edge ex   [E]
    float* m    = e + E;                 // seg max   [Npad]
    float* den  = m + Npad;              // seg sum   [Npad]
    float* sums = den + Npad;            // pooled    [G*64]
    float* cnt  = sums + (G << 6);       // counts    [G]
    (void)ws_size; (void)n_in;

    const int Mtiles = N / 16;                    // 3125
    const int mblk   = (Mtiles + MT - 1) / MT;    // 782
    const int Gtiles = G / 16;                    // 32
    const int gmblk  = (Gtiles + MT - 1) / MT;    // 8
    const int eblk  = (E * 32 + 255) / 256;
    const int nblk  = (N64 + 255) / 256;
    const int e1blk = (E + 255) / 256;
    const int pblk  = ((N * 32) + 255) / 256;
    const int gblk  = ((G << 6) + 255) / 256;

    // ---------------- Layer 1 ----------------
    gemm_wmma_f32<<<mblk, 128, 0, stream>>>(x, Wl1, bl1, buf0, Mtiles, 64);
    gemm_wmma_f32<<<mblk, 128, 0, stream>>>(x, Wr1, br1, buf1, Mtiles, 64);
    init_layer  <<<nblk, 256, 0, stream>>>(buf2, m, den, N);
    edge_scores <<<eblk, 256, 0, stream>>>(buf0, buf1, src, dst, att1, e, m, E);
    edge_exp    <<<e1blk, 256, 0, stream>>>(e, m, den, dst, E);
    edge_scatter<<<eblk, 256, 0, stream>>>(buf0, src, dst, e, den, buf2, E);
    bias_relu   <<<nblk, 256, 0, stream>>>(buf2, bias1, N64);

    // ---------------- Layer 2 ----------------
    gemm_wmma_f32<<<mblk, 128, 0, stream>>>(buf2, Wl2, bl2, buf0, Mtiles, 64);
    gemm_wmma_f32<<<mblk, 128, 0, stream>>>(buf2, Wr2, br2, buf1, Mtiles, 64);
    init_layer  <<<nblk, 256, 0, stream>>>(buf2, m, den, N);
    edge_scores <<<eblk, 256, 0, stream>>>(buf0, buf1, src, dst, att2, e, m, E);
    edge_exp    <<<e1blk, 256, 0, stream>>>(e, m, den, dst, E);
    edge_scatter<<<eblk, 256, 0, stream>>>(buf0, src, dst, e, den, buf2, E);
    bias_relu   <<<nblk, 256, 0, stream>>>(buf2, bias2, N64);

    // ---------------- Pool + classifier ----------------
    pool_init    <<<gblk, 256, 0, stream>>>(sums, cnt, G);
    pool_scatter <<<pblk, 256, 0, stream>>>(buf2, batch, sums, cnt, N);
    pool_finalize<<<gblk, 256, 0, stream>>>(sums, cnt, G);
    gemm_wmma_f32<<<gmblk, 128, 0, stream>>>(sums, lin_w, lin_b,
                                             (float*)d_out, Gtiles, 32);
}